// OMP_56573309224251
// MI455X (gfx1250) — compile-verified
//
#include <hip/hip_runtime.h>

typedef __attribute__((ext_vector_type(2))) float v2f;
typedef __attribute__((ext_vector_type(8))) float v8f;

#define N_DIM    2048
#define BATCH    2048
#define SPARSITY 16

static __device__ __forceinline__ v8f wmma4(v2f a, v2f b, v8f c) {
    return __builtin_amdgcn_wmma_f32_16x16x4_f32(
        false, a, false, b, (short)0, c, false, false);
}

// ---------------------------------------------------------------------------
// fp32 WMMA GEMM, all matrices 2048x2048 stride 2048: C = op(A)*op(B).
// Templated on transpose flags -> branch-free inner loop.
// Each wave computes a 32x32 tile (2x2 grid of 16x16 WMMA tiles): each K-step
// issues 4 independent v_wmma_f32_16x16x4_f32 with 2 operand floats per wmma.
// VGPR layouts per CDNA5 ISA 7.12.2:
//   A (16x4):  vgpr r: lanes 0-15 K=r, lanes 16-31 K=r+2; M = lane&15
//   B (4x16):  vgpr r: lanes 0-15 K=r, lanes 16-31 K=r+2; N = lane&15
//   C/D:       vgpr r: lanes 0-15 M=r, lanes 16-31 M=r+8; N = lane&15
// ---------------------------------------------------------------------------
template <bool TA, bool TB>
__global__ __launch_bounds__(256)
void wmma_gemm2048(const float* __restrict__ A, const float* __restrict__ B,
                   float* __restrict__ C)
{
    const int n    = N_DIM;
    const int wave = threadIdx.x >> 5;
    const int lane = threadIdx.x & 31;
    const int tile = blockIdx.x * 8 + wave;     // 8 waves / block
    const int tiles_x = n / 32;                 // 64
    const int ty = tile / tiles_x;
    const int tx = tile % tiles_x;
    const int m0 = ty * 32;
    const int n0 = tx * 32;

    const int half = lane >> 4;                 // 0 or 1
    const int l15  = lane & 15;
    const int am0  = m0 + l15;                  // A rows (two M tiles)
    const int am1  = am0 + 16;
    const int bn0  = n0 + l15;                  // B cols (two N tiles)
    const int bn1  = bn0 + 16;

    v8f c00 = {}, c01 = {}, c10 = {}, c11 = {};

#pragma unroll 2
    for (int k0 = 0; k0 < n; k0 += 4) {
        const int kA = k0 + 2 * half;           // this half-wave's K pair
        v2f a0, a1, b0, b1;
        if (TA) {                               // op(A)[m,k] = A[k*n+m]: column loads
            a0.x = A[(size_t)kA * n + am0];
            a0.y = A[(size_t)(kA + 1) * n + am0];
            a1.x = A[(size_t)kA * n + am1];
            a1.y = A[(size_t)(kA + 1) * n + am1];
        } else {                                // row-major: contiguous b64 load
            a0 = *(const v2f*)(A + (size_t)am0 * n + kA);
            a1 = *(const v2f*)(A + (size_t)am1 * n + kA);
        }
        if (TB) {                               // op(B)[k,j] = B[j*n+k]: contiguous b64
            b0 = *(const v2f*)(B + (size_t)bn0 * n + kA);
            b1 = *(const v2f*)(B + (size_t)bn1 * n + kA);
        } else {                                // row-major: column loads (coalesced)
            b0.x = B[(size_t)kA * n + bn0];
            b0.y = B[(size_t)(kA + 1) * n + bn0];
            b1.x = B[(size_t)kA * n + bn1];
            b1.y = B[(size_t)(kA + 1) * n + bn1];
        }
        c00 = wmma4(a0, b0, c00);
        c01 = wmma4(a0, b1, c01);
        c10 = wmma4(a1, b0, c10);
        c11 = wmma4(a1, b1, c11);
    }

    const int mb0 = m0 + half * 8;
    const int mb1 = mb0 + 16;
#pragma unroll
    for (int r = 0; r < 8; ++r) {
        C[(size_t)(mb0 + r) * n + n0 + l15]      = c00[r];
        C[(size_t)(mb0 + r) * n + n0 + 16 + l15] = c01[r];
        C[(size_t)(mb1 + r) * n + n0 + l15]      = c10[r];
        C[(size_t)(mb1 + r) * n + n0 + 16 + l15] = c11[r];
    }
}

// ---------------------------------------------------------------------------
// OMP greedy loop. One 256-thread block (8 wave32s) per batch element.
// Dynamic LDS caches this block's working set (fits in CDNA5's 320KB WGP LDS):
//   s_c0  [2048]          : C0 row for this batch element        ( 8 KB)
//   s_rows[16][2048]      : Mt rows of the selected atoms        (128 KB)
// Each Mt row is fetched from L2 exactly once (when selected); all later
// corr scans read LDS (conflict-free: lane i -> bank i).
//   corr_i = |s_c0[i] - sum_{j<t} x_j * s_rows[j][i]|
//   block argmax (first occurrence on ties), mask selected,
//   solve G[S,S] x = B0[b,S] via GE (SPD Gram -> no pivoting needed).
// ---------------------------------------------------------------------------
__global__ __launch_bounds__(256)
void omp_iterate(const float* __restrict__ Mt, const float* __restrict__ G,
                 const float* __restrict__ C0, const float* __restrict__ B0,
                 float* __restrict__ out)
{
    const int n   = N_DIM;
    const int b   = blockIdx.x;
    const int tid = threadIdx.x;

    extern __shared__ float smem[];
    float* s_rows = smem;                    // [SPARSITY][N_DIM]
    float* s_c0   = smem + SPARSITY * N_DIM; // [N_DIM]

    __shared__ float s_x[SPARSITY];
    __shared__ int   s_idx[SPARSITY];
    __shared__ float s_g[SPARSITY * SPARSITY];
    __shared__ float s_rhs[SPARSITY];
    __shared__ float s_lu[SPARSITY * (SPARSITY + 1)];
    __shared__ float s_val[256];
    __shared__ int   s_arg[256];

    // stage this batch element's C0 row into LDS (read 16x afterwards)
    const float* c0g = C0 + (size_t)b * n;
    for (int i = tid; i < n; i += 256) s_c0[i] = c0g[i];
    __syncthreads();

    for (int t = 0; t < SPARSITY; ++t) {
        // ---- fused corr + mask + local argmax (LDS-resident operands) -------
        float best  = -2.0f;
        int   besti = 0;
        for (int i = tid; i < n; i += 256) {
            float corr = s_c0[i];
            for (int j = 0; j < t; ++j)
                corr = fmaf(-s_x[j], s_rows[j * N_DIM + i], corr);
            corr = fabsf(corr);
            bool sel = false;
            for (int j = 0; j < t; ++j) sel |= (i == s_idx[j]);
            if (sel) corr = -1.0f;
            if (corr > best) { best = corr; besti = i; }   // keeps lowest i on tie
        }
        s_val[tid] = best;
        s_arg[tid] = besti;
        __syncthreads();
        // ---- block argmax reduction (prefer lower index on ties) ------------
        for (int s = 128; s > 0; s >>= 1) {
            if (tid < s) {
                const float v2 = s_val[tid + s];
                const int   a2 = s_arg[tid + s];
                if (v2 > s_val[tid] || (v2 == s_val[tid] && a2 < s_arg[tid])) {
                    s_val[tid] = v2;
                    s_arg[tid] = a2;
                }
            }
            __syncthreads();
        }
        if (tid == 0) s_idx[t] = s_arg[0];
        __syncthreads();
        const int it = s_idx[t];

        // ---- cache the newly selected Mt row in LDS (once, coalesced) -------
        if (t + 1 < SPARSITY) {
            const float* mrow = Mt + (size_t)it * n;
            for (int i = tid; i < n; i += 256) s_rows[t * N_DIM + i] = mrow[i];
        }

        // ---- extend Gram (G symmetric) and RHS ------------------------------
        if (tid <= t) {
            const float g = G[(size_t)it * n + s_idx[tid]];
            s_g[t * SPARSITY + tid] = g;
            s_g[tid * SPARSITY + t] = g;
        }
        if (tid == 0) s_rhs[t] = B0[(size_t)b * n + it];
        __syncthreads();

        // ---- (t+1)x(t+1) solve, serial GE by thread 0 -----------------------
        if (tid == 0) {
            const int m = t + 1;
            for (int r = 0; r < m; ++r) {
                for (int c = 0; c < m; ++c)
                    s_lu[r * (SPARSITY + 1) + c] = s_g[r * SPARSITY + c];
                s_lu[r * (SPARSITY + 1) + m] = s_rhs[r];
            }
            for (int p = 0; p < m; ++p) {
                const float inv = 1.0f / s_lu[p * (SPARSITY + 1) + p];
                for (int r = p + 1; r < m; ++r) {
                    const float f = s_lu[r * (SPARSITY + 1) + p] * inv;
                    for (int c = p; c <= m; ++c)
                        s_lu[r * (SPARSITY + 1) + c] -= f * s_lu[p * (SPARSITY + 1) + c];
                }
            }
            for (int r = m - 1; r >= 0; --r) {
                float v = s_lu[r * (SPARSITY + 1) + m];
                for (int c = r + 1; c < m; ++c)
                    v -= s_lu[r * (SPARSITY + 1) + c] * s_x[c];
                s_x[r] = v / s_lu[r * (SPARSITY + 1) + r];
            }
        }
        __syncthreads();
    }

    // ---- scatter sparse result (d_out is poisoned; zero the full row) -------
    float* orow = out + (size_t)b * n;
    for (int i = tid; i < n; i += 256) orow[i] = 0.0f;
    __syncthreads();
    if (tid < SPARSITY) orow[s_idx[tid]] = s_x[tid];
}

// ---------------------------------------------------------------------------
extern "C" void kernel_launch(void* const* d_in, const int* in_sizes, int n_in,
                              void* d_out, int out_size, void* d_ws, size_t ws_size,
                              hipStream_t stream)
{
    (void)in_sizes; (void)n_in; (void)out_size; (void)ws_size;
    const float* Y = (const float*)d_in[0];   // measurements (BATCH, N_DIM)
    const float* A = (const float*)d_in[1];   // dictionary  (N_DIM, N_DIM)
    float* out = (float*)d_out;

    const size_t NN = (size_t)N_DIM * N_DIM;
    float* Mt = (float*)d_ws;                 // (A*A)^T = A^T * A^T
    float* G  = Mt + NN;                      // A^T * A
    float* C0 = G  + NN;                      // Y * A^T  (row b = A*y_b)
    float* B0 = C0 + NN;                      // Y * A    (row b = A^T*y_b)

    const dim3 block(256);
    const dim3 grid((N_DIM / 32) * (N_DIM / 32) / 8);   // 32x32 tile per wave

    hipLaunchKernelGGL((wmma_gemm2048<true,  true >), grid, block, 0, stream, A, A, Mt);
    hipLaunchKernelGGL((wmma_gemm2048<true,  false>), grid, block, 0, stream, A, A, G);
    hipLaunchKernelGGL((wmma_gemm2048<false, true >), grid, block, 0, stream, Y, A, C0);
    hipLaunchKernelGGL((wmma_gemm2048<false, false>), grid, block, 0, stream, Y, A, B0);

    const size_t dynLds = (size_t)(SPARSITY * N_DIM + N_DIM) * sizeof(float); // 136 KB
    hipLaunchKernelGGL(omp_iterate, dim3(BATCH), dim3(256), dynLds, stream,
                       Mt, G, C0, B0, out);
}